// StateSpaceLayer_89558658056732
// MI455X (gfx1250) — compile-verified
//
#include <hip/hip_runtime.h>
#include <hip/hip_bf16.h>

// ---------------------------------------------------------------------------
// StateSpaceLayer for MI455X (gfx1250), f32 WMMA 16x16x4 everywhere.
//
//   dt   = exp(dt_log)
//   A_dT = (I - exp(A_log)*dt)^T     (B-operand of x_new = x_old @ A_dT)
//   Bu   = u @ (B*dt)^T              GEMM 16384x256, K=512
//   scan x_t = x_{t-1} @ A_dT + Bu_t chunked (T=64) parallel scan:
//              phase1 local scans -> Q=A_dT^64 boundary scan -> seeded rescan
//   out  = xs @ C^T + u @ D^T        GEMMs 16384x512, K=256/512
//
// CDNA5 features used: v_wmma_f32_16x16x4_f32, TDM tensor_load_to_lds to keep
// the 256KB A_dT resident in the 320KB WGP LDS, s_wait_tensorcnt, LDS ping-
// pong state tile, 4-way split accumulators to break WMMA RAW chains.
//
// All W ("B"-operand) matrices are stored pair-interleaved:
//   Wp[(k>>1)*2N + 2n + (k&1)] = W[k][n]   -> one b64 load per 4x16 B-tile.
// ---------------------------------------------------------------------------

typedef float        v2f  __attribute__((ext_vector_type(2)));
typedef float        v4f  __attribute__((ext_vector_type(4)));
typedef float        v8f  __attribute__((ext_vector_type(8)));
typedef unsigned int u32;
typedef u32          v4u  __attribute__((ext_vector_type(4)));
typedef int          v4i  __attribute__((ext_vector_type(4)));
typedef int          v8i  __attribute__((ext_vector_type(8)));

#define BB   8
#define LL   2048
#define IND  512
#define SD   256
#define OD   512
#define TCH  64            // chunk length
#define CPB  32            // chunks per batch
#define XS   258           // padded LDS row stride (floats)

// workspace layout (floats); total 9,437,184 floats = 37.75 MB
#define OFF_ADT  0u          // 256*256   A_dT plain (A-operand of squarings)
#define OFF_ADTP 65536u      // 256*256   A_dT packed (W-operand)
#define OFF_P1   131072u     // power ping  plain
#define OFF_P1P  196608u     // power ping  packed
#define OFF_P2   262144u     // power pong  plain
#define OFF_P2P  327680u     // power pong  packed  (Q ends here)
#define OFF_BT   393216u     // 512*256   (B*dt)^T packed
#define OFF_CT   524288u     // 256*512   C^T packed
#define OFF_DT   655360u     // 512*512   D^T packed
#define OFF_BU   917504u     // 16384*256 Bu (b,l,s)
#define OFF_LF   5111808u    // 256*256   local final states, row g=c*8+b
#define OFF_HS   5177344u    // 256*256   chunk start states, row g=c*8+b
#define OFF_XS   5242880u    // 16384*256 all states xs (b,l,s)

static __device__ __forceinline__ v8f wmma4(v2f a, v2f b, v8f c) {
  return __builtin_amdgcn_wmma_f32_16x16x4_f32(
      false, a, false, b, (short)0, c, false, false);
}

// ---------------------------------------------------------------------------
// TDM: DMA a 2D f32 tile (dim1 rows x dim0 elems, row stride = stride0 elems)
// from global memory into LDS at byte offset lds_off.  Issued by one wave;
// caller must s_wait_tensorcnt + barrier.  Descriptor per CDNA5 ISA ch.8.
// This toolchain's builtin takes 6 args (4 SGPR groups + spare + cpol).
// ---------------------------------------------------------------------------
static __device__ __forceinline__ void tdm_load_f32(const float* gsrc,
                                                    u32 lds_off,
                                                    u32 dim0, u32 dim1,
                                                    u32 stride0) {
  unsigned long long ga = (unsigned long long)(uintptr_t)gsrc;
  v4u g0;
  g0[0] = 1u;                                        // count=1, user desc
  g0[1] = lds_off;                                   // lds_addr (bytes)
  g0[2] = (u32)(ga & 0xffffffffu);                   // global_addr[31:0]
  g0[3] = (u32)((ga >> 32) & 0x01ffffffu) | (2u << 30); // addr[56:32] | type=2
  v8i g1;
  g1[0] = (int)(2u << 16);                           // data_size=4B
  g1[1] = (int)((dim0 & 0xffffu) << 16);             // tensor_dim0 lo16
  g1[2] = (int)(((dim0 >> 16) & 0xffffu) | ((dim1 & 0xffffu) << 16));
  g1[3] = (int)(((dim1 >> 16) & 0xffffu) | ((dim0 & 0xffffu) << 16)); // tile_dim0
  g1[4] = (int)(dim1 & 0xffffu);                     // tile_dim1 (tile_dim2=0)
  g1[5] = (int)stride0;                              // tensor_dim0_stride lo32
  g1[6] = 0;
  g1[7] = 0;
  v4i z4; z4[0] = 0; z4[1] = 0; z4[2] = 0; z4[3] = 0;  // 2D: groups 2/3 unused
  v8i z8;
#pragma unroll
  for (int i = 0; i < 8; ++i) z8[i] = 0;
  __builtin_amdgcn_tensor_load_to_lds(g0, g1, z4, z4, z8, 0);
}

// ---------------------------------------------------------------------------
// Prep: dt, A_dT (plain+packed), packed transposed weights.
// ---------------------------------------------------------------------------
__global__ void ssm_prep(const float* __restrict__ A_log,
                         const float* __restrict__ Bm,
                         const float* __restrict__ Cm,
                         const float* __restrict__ Dm,
                         const float* __restrict__ dt_log,
                         float* __restrict__ ws) {
  const float dt = expf(dt_log[0]);
  const int tid = blockIdx.x * blockDim.x + threadIdx.x;
  const int stride = gridDim.x * blockDim.x;
  for (int i = tid; i < SD * SD; i += stride) {       // A_dT[k][n] = A_d[n][k]
    int k = i >> 8, n = i & 255;
    float v = -expf(A_log[n * SD + k]) * dt;
    if (k == n) v += 1.0f;
    ws[OFF_ADT + i] = v;
    ws[OFF_ADTP + (size_t)(k >> 1) * 512 + (n << 1) + (k & 1)] = v;
  }
  for (int i = tid; i < IND * SD; i += stride) {      // (B*dt)^T packed
    int k = i >> 8, s = i & 255;
    ws[OFF_BT + (size_t)(k >> 1) * 512 + (s << 1) + (k & 1)] = Bm[s * IND + k] * dt;
  }
  for (int i = tid; i < SD * OD; i += stride) {       // C^T packed
    int s = i >> 9, o = i & 511;
    ws[OFF_CT + (size_t)(s >> 1) * 1024 + (o << 1) + (s & 1)] = Cm[o * SD + s];
  }
  for (int i = tid; i < IND * OD; i += stride) {      // D^T packed
    int k = i >> 9, o = i & 511;
    ws[OFF_DT + (size_t)(k >> 1) * 1024 + (o << 1) + (k & 1)] = Dm[o * IND + k];
  }
}

// ---------------------------------------------------------------------------
// f32 WMMA GEMM: Cout = A[M][K] @ W[K][N] (+Add).  W pair-interleaved.
// Block = 8 waves; a block owns (8/tilesN) M-tiles x all N; the A panel is
// staged in LDS once (dynamic shared = (8/tilesN)*16*K floats).
// Optional CoutP: also emit pair-interleaved copy (for the power chain).
// M%16==0, N in {256,512}, K%4==0.
// ---------------------------------------------------------------------------
__global__ void __launch_bounds__(256)
ssm_wmma_gemm(const float* __restrict__ A, int lda,
              const float* __restrict__ Wp,
              const float* __restrict__ Add,
              float* __restrict__ Cout, int ldc,
              float* __restrict__ CoutP,
              int M, int N, int K) {
  extern __shared__ float sA[];
  const int tid = threadIdx.x;
  const int tilesN = N >> 6;              // 4 or 8
  const int mtPerBlk = 8 / tilesN;        // 2 or 1
  const int rows = mtPerBlk << 4;
  const int mBase = blockIdx.x * rows;

  // cooperative A-panel staging (float4)
  for (int i = tid; i < (rows * K) >> 2; i += 256) {
    const int e = i << 2;
    const int r = e / K, k = e - r * K;
    *(v4f*)(sA + e) = *(const v4f*)(A + (size_t)(mBase + r) * lda + k);
  }
  __syncthreads();

  const int wib = tid >> 5, lane = tid & 31;
  const int lmt = wib / tilesN, nt = wib - lmt * tilesN;
  const int n0 = nt << 6;
  const int ln = lane & 15;
  const int hi = (lane >> 4) & 1;
  const int koff = hi << 1;
  const int m0 = mBase + (lmt << 4);
  const float* sArow = sA + (size_t)((lmt << 4) + ln) * K;
  const int N2 = N << 1;

  v8f acc[4];
  if (Add) {
#pragma unroll
    for (int j = 0; j < 4; ++j)
#pragma unroll
      for (int v = 0; v < 8; ++v)
        acc[j][v] = Add[(size_t)(m0 + v + (hi << 3)) * ldc + n0 + (j << 4) + ln];
  } else {
#pragma unroll
    for (int j = 0; j < 4; ++j) acc[j] = (v8f)(0.0f);
  }

  const int ksteps = K >> 2;
  for (int kk = 0; kk < ksteps; ++kk) {
    const int k0 = (kk << 2) + koff;
    const v2f a = *(const v2f*)(sArow + k0);
    const float* wrow = Wp + (size_t)(k0 >> 1) * N2;
#pragma unroll
    for (int j = 0; j < 4; ++j) {
      const v2f b = *(const v2f*)(wrow + ((n0 + (j << 4) + ln) << 1));
      acc[j] = wmma4(a, b, acc[j]);
    }
  }
#pragma unroll
  for (int j = 0; j < 4; ++j)
#pragma unroll
    for (int v = 0; v < 8; ++v) {
      const int rowm = m0 + v + (hi << 3);
      const int col = n0 + (j << 4) + ln;
      Cout[(size_t)rowm * ldc + col] = acc[j][v];
      if (CoutP)
        CoutP[(size_t)(rowm >> 1) * N2 + (col << 1) + (rowm & 1)] = acc[j][v];
    }
}

// ---------------------------------------------------------------------------
// Scan step helper: X(16x256) = X @ W + acc-init, 16 waves x 16 cols,
// W = packed 256x256 matrix resident in LDS, X ping-pong in LDS.
// 4 split accumulators break the WMMA RAW chain.
// ---------------------------------------------------------------------------
static __device__ __forceinline__ v8f scan_step(const float* __restrict__ Xc,
                                                const float* __restrict__ sWp,
                                                int ln, int n0, int koff,
                                                v8f acc0) {
  v8f acc1 = (v8f)(0.0f), acc2 = (v8f)(0.0f), acc3 = (v8f)(0.0f);
  const float* xrow = Xc + ln * XS;
  for (int kk = 0; kk < 64; kk += 4) {
    const int k0 = (kk << 2) + koff;
    v2f a0 = *(const v2f*)(xrow + k0);
    v2f a1 = *(const v2f*)(xrow + k0 + 4);
    v2f a2 = *(const v2f*)(xrow + k0 + 8);
    v2f a3 = *(const v2f*)(xrow + k0 + 12);
    v2f b0 = *(const v2f*)(sWp + (size_t)((k0)      >> 1) * 512 + ((n0 + ln) << 1));
    v2f b1 = *(const v2f*)(sWp + (size_t)((k0 + 4)  >> 1) * 512 + ((n0 + ln) << 1));
    v2f b2 = *(const v2f*)(sWp + (size_t)((k0 + 8)  >> 1) * 512 + ((n0 + ln) << 1));
    v2f b3 = *(const v2f*)(sWp + (size_t)((k0 + 12) >> 1) * 512 + ((n0 + ln) << 1));
    acc0 = wmma4(a0, b0, acc0);
    acc1 = wmma4(a1, b1, acc1);
    acc2 = wmma4(a2, b2, acc2);
    acc3 = wmma4(a3, b3, acc3);
  }
  return (acc0 + acc1) + (acc2 + acc3);
}

// ---------------------------------------------------------------------------
// Phase 1: local scans from zero state. 16 WGs x 16 rows (row g = c*8+b).
// ---------------------------------------------------------------------------
__global__ void __launch_bounds__(512)
ssm_scan_phase1(const float* __restrict__ Adtp,
                const float* __restrict__ Bu,
                float* __restrict__ LF) {
  __shared__ float sWp[SD * SD];          // 256 KB packed A_dT, TDM-resident
  __shared__ float sX[2][16 * XS];        // 33 KB state ping-pong
  const int tid = threadIdx.x;
  if ((tid >> 5) == 0) {                  // wave 0 drives the tensor DMA
    tdm_load_f32(Adtp, (u32)(uintptr_t)&sWp[0], 16384u, 4u, 16384u);
    __builtin_amdgcn_s_wait_tensorcnt(0);
  }
  for (int i = tid; i < 16 * XS; i += 512) { sX[0][i] = 0.0f; sX[1][i] = 0.0f; }
  __syncthreads();

  const int wave = tid >> 5, lane = tid & 31;
  const int ln = lane & 15, hi = (lane >> 4) & 1, koff = hi << 1;
  const int n0 = wave << 4;
  const int gbase = blockIdx.x << 4;

  v8f acc = (v8f)(0.0f);
  for (int t = 0; t < TCH; ++t) {
    const float* Xc = sX[t & 1];
    float* Xn = sX[(t & 1) ^ 1];
#pragma unroll
    for (int v = 0; v < 8; ++v) {          // acc-init := Bu tile at time t
      const int g = gbase + v + (hi << 3);
      const int b = g & 7, c = g >> 3;
      acc[v] = Bu[(size_t)(((b << 11) + (c << 6) + t) << 8) + n0 + ln];
    }
    acc = scan_step(Xc, sWp, ln, n0, koff, acc);
    __syncthreads();
#pragma unroll
    for (int v = 0; v < 8; ++v) Xn[(v + (hi << 3)) * XS + n0 + ln] = acc[v];
    __syncthreads();
  }
#pragma unroll
  for (int v = 0; v < 8; ++v)
    LF[(size_t)(gbase + v + (hi << 3)) * SD + n0 + ln] = acc[v];
}

// ---------------------------------------------------------------------------
// Phase 2: boundary scan (one WG): H_{c+1} = H_c @ Q + LF_c, Q = A_dT^64.
// ---------------------------------------------------------------------------
__global__ void __launch_bounds__(512)
ssm_chunk_scan(const float* __restrict__ state,
               const float* __restrict__ Qp,
               const float* __restrict__ LF,
               float* __restrict__ Hstart,
               float* __restrict__ outTail) {
  __shared__ float sWp[SD * SD];
  __shared__ float sX[2][16 * XS];
  const int tid = threadIdx.x;
  if ((tid >> 5) == 0) {
    tdm_load_f32(Qp, (u32)(uintptr_t)&sWp[0], 16384u, 4u, 16384u);
    __builtin_amdgcn_s_wait_tensorcnt(0);
  }
  for (int i = tid; i < 16 * XS; i += 512) {
    const int r = i / XS, cc = i - r * XS;
    sX[0][i] = (r < 8 && cc < SD) ? state[(r << 8) + cc] : 0.0f;
    sX[1][i] = 0.0f;
  }
  for (int i = tid; i < 8 * SD; i += 512) Hstart[i] = state[i];   // chunk 0
  __syncthreads();

  const int wave = tid >> 5, lane = tid & 31;
  const int ln = lane & 15, hi = (lane >> 4) & 1, koff = hi << 1;
  const int n0 = wave << 4;

  for (int c = 0; c < CPB; ++c) {
    const float* Xc = sX[c & 1];
    float* Xn = sX[(c & 1) ^ 1];
    v8f acc;
#pragma unroll
    for (int v = 0; v < 8; ++v) {
      const int rowm = v + (hi << 3);
      acc[v] = (rowm < 8) ? LF[(size_t)((c << 3) + rowm) * SD + n0 + ln] : 0.0f;
    }
    acc = scan_step(Xc, sWp, ln, n0, koff, acc);
    __syncthreads();
#pragma unroll
    for (int v = 0; v < 8; ++v) Xn[(v + (hi << 3)) * XS + n0 + ln] = acc[v];
#pragma unroll
    for (int v = 0; v < 8; ++v) {
      const int rowm = v + (hi << 3);
      if (rowm < 8) {
        if (c < CPB - 1)
          Hstart[(size_t)(((c + 1) << 3) + rowm) * SD + n0 + ln] = acc[v];
        else
          outTail[(rowm << 8) + n0 + ln] = acc[v];  // final_state
      }
    }
    __syncthreads();
  }
}

// ---------------------------------------------------------------------------
// Phase 3: rescan seeded with true chunk-start states; emit all states xs.
// ---------------------------------------------------------------------------
__global__ void __launch_bounds__(512)
ssm_scan_phase3(const float* __restrict__ Adtp,
                const float* __restrict__ Bu,
                const float* __restrict__ Hstart,
                float* __restrict__ xs) {
  __shared__ float sWp[SD * SD];
  __shared__ float sX[2][16 * XS];
  const int tid = threadIdx.x;
  const int gbase = blockIdx.x << 4;
  if ((tid >> 5) == 0) {
    tdm_load_f32(Adtp, (u32)(uintptr_t)&sWp[0], 16384u, 4u, 16384u);
    __builtin_amdgcn_s_wait_tensorcnt(0);
  }
  for (int i = tid; i < 16 * XS; i += 512) {
    const int r = i / XS, cc = i - r * XS;
    sX[0][i] = (cc < SD) ? Hstart[(size_t)(gbase + r) * SD + cc] : 0.0f;
    sX[1][i] = 0.0f;
  }
  __syncthreads();

  const int wave = tid >> 5, lane = tid & 31;
  const int ln = lane & 15, hi = (lane >> 4) & 1, koff = hi << 1;
  const int n0 = wave << 4;

  for (int t = 0; t < TCH; ++t) {
    const float* Xc = sX[t & 1];
    float* Xn = sX[(t & 1) ^ 1];
    v8f acc;
#pragma unroll
    for (int v = 0; v < 8; ++v) {
      const int g = gbase + v + (hi << 3);
      const int b = g & 7, c = g >> 3;
      acc[v] = Bu[(size_t)(((b << 11) + (c << 6) + t) << 8) + n0 + ln];
    }
    acc = scan_step(Xc, sWp, ln, n0, koff, acc);
    __syncthreads();
#pragma unroll
    for (int v = 0; v < 8; ++v) Xn[(v + (hi << 3)) * XS + n0 + ln] = acc[v];
#pragma unroll
    for (int v = 0; v < 8; ++v) {
      const int g = gbase + v + (hi << 3);
      const int b = g & 7, c = g >> 3;
      xs[(size_t)(((b << 11) + (c << 6) + t) << 8) + n0 + ln] = acc[v];
    }
    __syncthreads();
  }
}

// ---------------------------------------------------------------------------
extern "C" void kernel_launch(void* const* d_in, const int* in_sizes, int n_in,
                              void* d_out, int out_size, void* d_ws, size_t ws_size,
                              hipStream_t stream) {
  const float* u      = (const float*)d_in[0];
  const float* state  = (const float*)d_in[1];
  const float* A_log  = (const float*)d_in[2];
  const float* Bm     = (const float*)d_in[3];
  const float* Cm     = (const float*)d_in[4];
  const float* Dm     = (const float*)d_in[5];
  const float* dt_log = (const float*)d_in[6];
  float* ws  = (float*)d_ws;                     // needs ~38 MB
  float* out = (float*)d_out;

  (void)in_sizes; (void)n_in; (void)out_size; (void)ws_size;

  ssm_prep<<<256, 256, 0, stream>>>(A_log, Bm, Cm, Dm, dt_log, ws);

  // Bu = u @ (B*dt)^T : M=16384, N=256, K=512; 32 rows/block -> 512 blocks
  ssm_wmma_gemm<<<512, 256, 32 * 512 * 4, stream>>>(
      u, IND, ws + OFF_BT, nullptr, ws + OFF_BU, SD, nullptr,
      BB * LL, SD, IND);

  // Q = A_dT^64 via 6 squarings (plain+packed ping/pong)
  const float* srcA = ws + OFF_ADT;
  const float* srcP = ws + OFF_ADTP;
  float* dA[2] = { ws + OFF_P1, ws + OFF_P2 };
  float* dP[2] = { ws + OFF_P1P, ws + OFF_P2P };
  for (int i = 0; i < 6; ++i) {
    ssm_wmma_gemm<<<8, 256, 32 * 256 * 4, stream>>>(
        srcA, SD, srcP, nullptr, dA[i & 1], SD, dP[i & 1], SD, SD, SD);
    srcA = dA[i & 1];
    srcP = dP[i & 1];                            // ends at OFF_P2P
  }

  // chunked scan
  ssm_scan_phase1<<<16, 512, 0, stream>>>(ws + OFF_ADTP, ws + OFF_BU, ws + OFF_LF);
  ssm_chunk_scan<<<1, 512, 0, stream>>>(state, srcP, ws + OFF_LF, ws + OFF_HS,
                                        out + (size_t)BB * LL * OD);
  ssm_scan_phase3<<<16, 512, 0, stream>>>(ws + OFF_ADTP, ws + OFF_BU,
                                          ws + OFF_HS, ws + OFF_XS);

  // out = xs @ C^T ; out += u @ D^T : M=16384, N=512 -> 1024 blocks each
  ssm_wmma_gemm<<<1024, 256, 16 * 256 * 4, stream>>>(
      ws + OFF_XS, SD, ws + OFF_CT, nullptr, out, OD, nullptr,
      BB * LL, OD, SD);
  ssm_wmma_gemm<<<1024, 256, 16 * 512 * 4, stream>>>(
      u, IND, ws + OFF_DT, out, out, OD, nullptr,
      BB * LL, OD, IND);
}